// DataEmbedding_26980984554070
// MI455X (gfx1250) — compile-verified
//
#include <hip/hip_runtime.h>
#include <hip/hip_bf16.h>

// ---- problem constants (from reference) ----
#define B_    16
#define S_    4096
#define CIN_  7
#define DM_   512
#define KERN_ 73          // D_MODEL // CIN
#define NLAG_ 6           // M+1
#define TAO_  3
#define MASK_T_ 15        // M*TAO: t < 15 -> lag values zeroed
#define TB_   128         // time steps per block (8 waves x 16)
// div[m] = exp(2m * (-ln(10000)/512)) -> exponent coefficient per m:
#define C0_  (-0.03597789207803197f)   // -ln(10000)/256
#define RROWS_ 32                       // temporal row table rows (idx < 32)

typedef float v2f __attribute__((ext_vector_type(2)));
typedef float v8f __attribute__((ext_vector_type(8)));

// ---------------------------------------------------------------------------
// Kernel 1: build (a) shared sinusoid-row table R[p][d], p<32, d<512
// (all four temporal tables share identical row contents; x_mark in [0,7)),
// and (b) divtab[m] = exp(m * C0), m<256, so the hot kernel has no expf.
// ---------------------------------------------------------------------------
__global__ void build_tabs_kernel(float* __restrict__ Rtab,
                                  float* __restrict__ divtab) {
    int j = blockIdx.x * 256 + threadIdx.x;
    if (j < RROWS_ * DM_) {
        int p = j / DM_, d = j % DM_;
        float div = expf((float)(d >> 1) * C0_);
        float ang = (float)p * div;
        Rtab[j] = (d & 1) ? cosf(ang) : sinf(ang);
    } else if (j < RROWS_ * DM_ + 256) {
        int m = j - RROWS_ * DM_;
        divtab[m] = expf((float)m * C0_);
    }
}

// masked, circular-wrapped lag-conv input value:
// value = ml_i((t-1+k) mod S), ml_i(w) = (w>=15) ? x[b, w-3i, c] : 0
__device__ __forceinline__ float lag_val(const float* __restrict__ x,
                                         const float* __restrict__ xseg,
                                         int b, int c, int t0, int t, int kk) {
    if (kk >= 18) return 0.f;                 // K padding 18 -> 20
    int i = kk / 3;
    int k = kk - 3 * i;
    int w = t - 1 + k;                        // unwrapped position in lag array
    if (w < 0) {                              // only w==-1: wraps to S-1 (>=15)
        int g = (w + S_) - 3 * i;             // S-1-3i >= 0
        return x[((size_t)b * S_ + g) * CIN_ + c];
    }
    if (w >= S_) return 0.f;                  // wraps to 0..2 < 15 -> masked 0
    if (w < MASK_T_) return 0.f;              // lag mask
    return xseg[w - 3 * i - (t0 - 16)];       // in-segment (j in [0,144])
}

// ---------------------------------------------------------------------------
// Kernel 2: WMMA lag-conv for channels d = c*73 + o (o in [0,73)), fused with
// bias + positional encoding + temporal embedding. One block per (t0, c, b):
// 8 waves, each owning a 16-row time tile; fully unrolled 5 o-tiles x 5
// K-steps of v_wmma_f32_16x16x4_f32. PE uses one sincos + angle-addition
// recurrence per tile instead of 8 sincos calls.
// ---------------------------------------------------------------------------
__global__ __launch_bounds__(256) void conv_wmma_kernel(
    const float* __restrict__ x,          // [B,S,7]
    const int*   __restrict__ x_mark,     // [B,S,5]
    const float* __restrict__ conv_w,     // [73,6,3]
    const float* __restrict__ conv_b,     // [73]
    const float* __restrict__ Rtab,       // [32,512]
    const float* __restrict__ divtab,     // [256]
    float*       __restrict__ out)        // [B,S,512]
{
    __shared__ float xseg[160];           // x[b, t0-16 .. t0+143, c] (masked OOR->0)
    __shared__ float Wl[80 * 20];         // weights padded: o<80, kk<20
    __shared__ int   mk[TB_ * 4];         // x_mark[b, t0..t0+127, 0..3]

    const int t0  = blockIdx.x * TB_;
    const int c   = blockIdx.y;
    const int b   = blockIdx.z;
    const int tid = threadIdx.x;

    // stage weights (zero-padded)
    for (int j = tid; j < 80 * 20; j += 256) {
        int o = j / 20, kk = j % 20;
        Wl[j] = (o < KERN_ && kk < 18) ? conv_w[o * 18 + kk] : 0.f;
    }
    // stage x segment
    for (int j = tid; j < 160; j += 256) {
        int g = t0 - 16 + j;
        xseg[j] = (g >= 0 && g < S_) ? x[((size_t)b * S_ + g) * CIN_ + c] : 0.f;
    }
    // stage temporal indices
    for (int j = tid; j < TB_ * 4; j += 256) {
        int t = t0 + (j >> 2), q = j & 3;
        mk[j] = x_mark[((size_t)b * S_ + t) * 5 + q];
    }
    __syncthreads();

    const int wave = tid >> 5;
    const int lane = tid & 31;
    const int tt0  = t0 + wave * 16;      // this wave's 16-row time tile
    const int nrow = lane & 15;           // A: row M / B: col N / C: col N
    const int hi   = lane >> 4;           // K half-select for A/B fragments

    // A fragments: 5 K-steps of 16x4, reused across all o-tiles
    v2f afrag[5];
#pragma unroll
    for (int s = 0; s < 5; ++s) {
        int kb = s * 4 + hi * 2;
        int t  = tt0 + nrow;
        afrag[s].x = lag_val(x, xseg, b, c, t0, t, kb);
        afrag[s].y = lag_val(x, xseg, b, c, t0, t, kb + 1);
    }

#pragma unroll
    for (int ot = 0; ot < 5; ++ot) {
        const int obase = ot * 16;
        v8f acc = {};
#pragma unroll
        for (int s = 0; s < 5; ++s) {
            int kb = s * 4 + hi * 2;
            v2f bfrag;
            bfrag.x = Wl[(obase + nrow) * 20 + kb];
            bfrag.y = Wl[(obase + nrow) * 20 + kb + 1];
            acc = __builtin_amdgcn_wmma_f32_16x16x4_f32(
                false, afrag[s], false, bfrag, (short)0, acc, false, false);
        }
        // fused epilogue: bias + PE + temporal, single store per element
        const int o = obase + nrow;
        if (o < KERN_) {
            const int   d    = c * KERN_ + o;
            const float bias = conv_b[o];
            const float div  = divtab[d >> 1];
            // base PE angle for this lane's 8 rows (t = tt0 + hi*8 + r),
            // advanced by angle-addition each step
            float sd, cd, s0, cc0;
            __sincosf(div, &sd, &cd);                 // step rotation
            float ang0 = (float)(tt0 + hi * 8) * div;
            sincosf(ang0, &s0, &cc0);                 // precise base
#pragma unroll
            for (int r = 0; r < 8; ++r) {
                int t = tt0 + hi * 8 + r;             // C/D layout: lanes16-31 -> M+8
                float pe = (d & 1) ? cc0 : s0;
                const int* mr = &mk[(t - t0) * 4];
                float tmp = Rtab[mr[0] * DM_ + d] + Rtab[mr[1] * DM_ + d]
                          + Rtab[mr[2] * DM_ + d] + Rtab[mr[3] * DM_ + d];
                out[((size_t)b * S_ + t) * DM_ + d] = acc[r] + bias + pe + tmp;
                // rotate (s0, c0) by div
                float sn = s0 * cd + cc0 * sd;
                float cn = cc0 * cd - s0 * sd;
                s0 = sn; cc0 = cn;
            }
        }
    }
}

// ---------------------------------------------------------------------------
// Kernel 3: remainder channel d=511 (left conv on last input channel's lags),
// fused with PE + temporal. One thread per (b,t).
// ---------------------------------------------------------------------------
__global__ void remainder_kernel(
    const float* __restrict__ x,
    const int*   __restrict__ x_mark,
    const float* __restrict__ left_w,     // [1,6,3]
    const float* __restrict__ left_b,     // [1]
    const float* __restrict__ Rtab,
    const float* __restrict__ divtab,
    float*       __restrict__ out)
{
    int j = blockIdx.x * 256 + threadIdx.x;   // B*S
    if (j >= B_ * S_) return;
    int b = j / S_, t = j % S_;

    float acc = left_b[0];
#pragma unroll
    for (int i = 0; i < NLAG_; ++i)
#pragma unroll
        for (int k = 0; k < 3; ++k) {
            int w = t - 1 + k;
            w = (w < 0) ? w + S_ : ((w >= S_) ? w - S_ : w);
            float v = (w >= MASK_T_)
                    ? x[((size_t)b * S_ + (w - 3 * i)) * CIN_ + (CIN_ - 1)]
                    : 0.f;
            acc += v * left_w[i * 3 + k];
        }

    const int d = DM_ - 1;                    // 511 (odd -> cos)
    float div = divtab[d >> 1];
    float pe  = cosf((float)t * div);
    const int* mr = &x_mark[(size_t)j * 5];
    float tmp = Rtab[mr[0] * DM_ + d] + Rtab[mr[1] * DM_ + d]
              + Rtab[mr[2] * DM_ + d] + Rtab[mr[3] * DM_ + d];
    out[(size_t)j * DM_ + d] = acc + pe + tmp;
}

// ---------------------------------------------------------------------------
extern "C" void kernel_launch(void* const* d_in, const int* in_sizes, int n_in,
                              void* d_out, int out_size, void* d_ws, size_t ws_size,
                              hipStream_t stream) {
    (void)in_sizes; (void)n_in; (void)out_size; (void)ws_size;
    const float* x      = (const float*)d_in[0];
    const int*   x_mark = (const int*)  d_in[1];
    const float* conv_w = (const float*)d_in[2];
    const float* conv_b = (const float*)d_in[3];
    const float* left_w = (const float*)d_in[4];
    const float* left_b = (const float*)d_in[5];
    float* out    = (float*)d_out;
    float* Rtab   = (float*)d_ws;                    // 32*512 floats = 64 KB
    float* divtab = Rtab + RROWS_ * DM_;             // +256 floats = 1 KB

    int tab_n = RROWS_ * DM_ + 256;
    build_tabs_kernel<<<(tab_n + 255) / 256, 256, 0, stream>>>(Rtab, divtab);

    dim3 grid(S_ / TB_, CIN_, B_);                   // (32, 7, 16)
    conv_wmma_kernel<<<grid, 256, 0, stream>>>(
        x, x_mark, conv_w, conv_b, Rtab, divtab, out);

    remainder_kernel<<<(B_ * S_ + 255) / 256, 256, 0, stream>>>(
        x, x_mark, left_w, left_b, Rtab, divtab, out);
}